// PSMCosineLayer_57853209477697
// MI455X (gfx1250) — compile-verified
//
#include <hip/hip_runtime.h>

typedef float v2f __attribute__((ext_vector_type(2)));
typedef float v4f __attribute__((ext_vector_type(4)));
typedef float v8f __attribute__((ext_vector_type(8)));

#define B_ 8
#define C_ 128
#define H_ 96
#define W_ 320
#define D_ 48
#define TW 16          // output w-tile width
#define RSTRIDE 80     // 64 staged right columns, padded to 80 (bank-conflict free)
#define GSTRIDE 20     // 16x16 G tile rows padded to 20 (bank-conflict free)

__global__ __launch_bounds__(128)
void psm_cost_volume_wmma(const float* __restrict__ L,
                          const float* __restrict__ R,
                          float* __restrict__ out)
{
    __shared__ float Lsh[C_ * TW];           // [c][i], row stride 16
    __shared__ float Rsh[C_ * RSTRIDE];      // [c][q], q in [0,64), row stride 80
    __shared__ float Gsh[4 * 16 * GSTRIDE];  // [s][m][n], row stride 20

    const int tid = threadIdx.x;
    const int b   = blockIdx.z;
    const int h   = blockIdx.y;
    const int w0  = blockIdx.x * TW;

    const size_t HW = (size_t)H_ * W_;
    const float* Lbase = L + ((size_t)b * C_) * HW + (size_t)h * W_ + w0;
    const float* Rbase = R + ((size_t)b * C_) * HW + (size_t)h * W_;

    // ---- stage left tile: 128 ch x 16 cols, coalesced b128, NON-TEMPORAL ----
    // (each L element read exactly once; keep it out of L2 so R stays resident)
#pragma unroll
    for (int k = 0; k < 4; ++k) {
        int e  = tid + 128 * k;      // 512 float4s
        int c  = e >> 2;
        int i4 = (e & 3) * 4;
        v4f v = __builtin_nontemporal_load((const v4f*)(Lbase + (size_t)c * HW + i4));
        *(v4f*)(&Lsh[c * TW + i4]) = v;
    }
    // ---- stage right cols [w0-48, w0+16), zero-fill w' < 0, REGULAR temporal ----
    // (R window overlaps between adjacent w-tiles -> want these lines in L2)
#pragma unroll
    for (int k = 0; k < 16; ++k) {
        int e  = tid + 128 * k;      // 2048 float4s
        int c  = e >> 4;
        int q4 = (e & 15) * 4;
        int wq = w0 - 48 + q4;       // multiple of 4: whole float4 in or out
        v4f v = {0.f, 0.f, 0.f, 0.f};
        if (wq >= 0)
            v = *(const v4f*)(Rbase + (size_t)c * HW + wq);
        *(v4f*)(&Rsh[c * RSTRIDE + q4]) = v;
    }
    __syncthreads();

    // ---- each wave computes Gram tile s: G_s[i][j] = dot(L[:,w0+i], R[:,w0-48+16s+j]) ----
    const int s    = tid >> 5;       // wave id 0..3
    const int lane = tid & 31;
    const int ln   = lane & 15;      // M (A) / N (B)
    const int hi   = lane >> 4;      // K-half select per ISA fp32 layout
    const int bcol = 16 * s + ln;

    v8f acc = {};
#pragma unroll
    for (int kc = 0; kc < 32; ++kc) {
        const int k0 = 4 * kc + 2 * hi;
        v2f a, bb;
        a.x  = Lsh[(k0 + 0) * TW + ln];
        a.y  = Lsh[(k0 + 1) * TW + ln];
        bb.x = Rsh[(k0 + 0) * RSTRIDE + bcol];
        bb.y = Rsh[(k0 + 1) * RSTRIDE + bcol];
        // 8 args: (neg_a, A, neg_b, B, c_mod, C, reuse_a, reuse_b)
        acc = __builtin_amdgcn_wmma_f32_16x16x4_f32(false, a, false, bb,
                                                    (short)0, acc, false, false);
    }

    // ---- spill C/D tile: VGPR r holds (M = r + 8*hi, N = ln) ----
#pragma unroll
    for (int r = 0; r < 8; ++r)
        Gsh[s * (16 * GSTRIDE) + (r + 8 * hi) * GSTRIDE + ln] = acc[r];
    __syncthreads();

    // ---- assemble band diagonals: out[b,d,h,w0+j] = G[w0+j, w0+j-d] / 128 ----
    // output is write-once streaming -> non-temporal stores
    float* obase = out + ((size_t)b * D_) * HW + (size_t)h * W_ + w0;
#pragma unroll
    for (int k = 0; k < 6; ++k) {
        int e  = tid + 128 * k;      // 768 outputs: 48 d x 16 j
        int d  = e >> 4;
        int j  = e & 15;
        int tp = d >> 4;
        int ip = d & 15;
        int ss, jj;                  // q = j - d + 48 = 16*ss + jj
        if (j >= ip) { ss = 3 - tp; jj = j - ip; }
        else         { ss = 2 - tp; jj = j - ip + 16; }
        float v = Gsh[ss * (16 * GSTRIDE) + j * GSTRIDE + jj] * (1.0f / C_);
        if (w0 + j < d) v = 0.0f;    // left zero-pad region
        __builtin_nontemporal_store(v, &obase[(size_t)d * HW + j]);
    }
}

extern "C" void kernel_launch(void* const* d_in, const int* in_sizes, int n_in,
                              void* d_out, int out_size, void* d_ws, size_t ws_size,
                              hipStream_t stream) {
    const float* L = (const float*)d_in[0];
    const float* R = (const float*)d_in[1];
    float* o = (float*)d_out;
    dim3 grid(W_ / TW, H_, B_);   // 20 x 96 x 8 workgroups
    psm_cost_volume_wmma<<<grid, 128, 0, stream>>>(L, R, o);
}